// CenterNetLoss_28071906246702
// MI455X (gfx1250) — compile-verified
//
#include <hip/hip_runtime.h>
#include <hip/hip_bf16.h>
#include <stdint.h>

#define EPSF 1e-4f

#define B_   16
#define C_   80
#define H_   128
#define W_   128
#define K_   128
#define HW_  (H_ * W_)
#define HMAP_ELEMS (B_ * C_ * HW_)      /* 20,971,520 */
#define HMAP_V4    (HMAP_ELEMS / 4)     /* 5,242,880  */
#define FOCAL_BLOCK 256
#define FOCAL_GRID  2048
#define NTILES (HMAP_V4 / FOCAL_BLOCK)  /* 20,480 ; divisible by FOCAL_GRID */

// ---------------------------------------------------------------------------
// Workspace layout (floats):
//  [0..2]   pos_loss  (tl, br, ct)
//  [3..5]   neg_loss
//  [6..8]   num_pos
//  [9..11]  smooth-L1 masked sums (tl, br, ct)
//  [12]     total mask sum
//  [13]     pull accumulator
//  [14]     push accumulator
//  [16..31] per-batch mask counts
//  [32 .. 32+2048)        gathered e_tl
//  [32+2048 .. 32+4096)   gathered e_br
// ---------------------------------------------------------------------------

// ---- CDNA5 async-to-LDS primitives (gfx1250) ------------------------------
__device__ __forceinline__ void async_load_b128(uint32_t lds_off, uint32_t voff,
                                                const float* base) {
  // GVS form: copies 16B per lane from global base saddr plus signed offset vaddr to LDS address in vdst.
  asm volatile("global_load_async_to_lds_b128 %0, %1, %2"
               :: "v"(lds_off), "v"(voff), "s"(base)
               : "memory");
}
__device__ __forceinline__ void wait_async_le6() {
  asm volatile("s_wait_asynccnt 6" ::: "memory");
}
__device__ __forceinline__ void wait_async_0() {
  asm volatile("s_wait_asynccnt 0" ::: "memory");
}

// ---- helpers --------------------------------------------------------------
__device__ __forceinline__ float wave_sum(float v) {
#pragma unroll
  for (int off = 16; off > 0; off >>= 1) v += __shfl_xor(v, off, 32);
  return v;
}

__device__ __forceinline__ void focal_step(float x, float y, float& pos,
                                           float& neg, float& cnt) {
  // p = clip(sigmoid(x), eps, 1-eps); 3 trans ops total (exp, rcp, log).
  float p = __builtin_amdgcn_rcpf(1.0f + __expf(-x));
  p = fminf(fmaxf(p, EPSF), 1.0f - EPSF);
  bool ispos = (y == 1.0f);
  float q   = ispos ? p : (1.0f - p);
  float omy = 1.0f - y;
  float o2  = omy * omy;
  float omp = 1.0f - p;
  float w   = ispos ? omp * omp : p * p * o2 * o2;
  float c   = __logf(q) * w;
  pos += ispos ? c : 0.0f;
  neg += ispos ? 0.0f : c;
  cnt += ispos ? 1.0f : 0.0f;
}

// ---- Kernel 1 (defined FIRST so its disasm shows in the snippet):
//      focal loss over 3 heatmap pairs, streamed via async-to-LDS pipeline --
__global__ __launch_bounds__(FOCAL_BLOCK) void focal_kernel(
    const float* __restrict__ p0, const float* __restrict__ p1,
    const float* __restrict__ p2, const float* __restrict__ g0,
    const float* __restrict__ g1, const float* __restrict__ g2,
    float* __restrict__ ws) {
  // [buffer][wave][array][lane] ; per-wave private regions -> no barriers,
  // only per-wave ASYNCcnt waits. 2*8*6*32*16B = 48 KiB LDS.
  __shared__ float4 smem[2][FOCAL_BLOCK / 32][6][32];

  const int tid = threadIdx.x;
  const int wv  = tid >> 5;
  const int ln  = tid & 31;

  const float* basep[3] = {p0, p1, p2};
  const float* baseg[3] = {g0, g1, g2};

  uint32_t lds[2][6];
#pragma unroll
  for (int bb = 0; bb < 2; ++bb)
#pragma unroll
    for (int a = 0; a < 6; ++a)
      lds[bb][a] = (uint32_t)(uintptr_t)&smem[bb][wv][a][ln];

  float posA[3] = {0.f, 0.f, 0.f};
  float negA[3] = {0.f, 0.f, 0.f};
  float cntA[3] = {0.f, 0.f, 0.f};

  int tile = (int)blockIdx.x;
  int buf  = 0;

  // Prologue: stage first tile (6 async b128 ops per lane = 96 B/lane).
  {
    uint32_t voff = (uint32_t)(tile * FOCAL_BLOCK + tid) * 16u;
#pragma unroll
    for (int a = 0; a < 3; ++a) {
      async_load_b128(lds[0][a],     voff, basep[a]);
      async_load_b128(lds[0][3 + a], voff, baseg[a]);
    }
  }

  for (; tile < NTILES; tile += FOCAL_GRID) {
    const int nxt = tile + FOCAL_GRID;
    if (nxt < NTILES) {
      // Stage next tile into the other buffer, then wait for the 6 older
      // transfers (allow the 6 newest to remain in flight).
      uint32_t voff = (uint32_t)(nxt * FOCAL_BLOCK + tid) * 16u;
#pragma unroll
      for (int a = 0; a < 3; ++a) {
        async_load_b128(lds[buf ^ 1][a],     voff, basep[a]);
        async_load_b128(lds[buf ^ 1][3 + a], voff, baseg[a]);
      }
      wait_async_le6();
    } else {
      wait_async_0();
    }

#pragma unroll
    for (int a = 0; a < 3; ++a) {
      float4 x = smem[buf][wv][a][ln];
      float4 y = smem[buf][wv][3 + a][ln];
      focal_step(x.x, y.x, posA[a], negA[a], cntA[a]);
      focal_step(x.y, y.y, posA[a], negA[a], cntA[a]);
      focal_step(x.z, y.z, posA[a], negA[a], cntA[a]);
      focal_step(x.w, y.w, posA[a], negA[a], cntA[a]);
    }
    buf ^= 1;
  }

#pragma unroll
  for (int a = 0; a < 3; ++a) {
    float s;
    s = wave_sum(posA[a]); if (ln == 0) atomicAdd(&ws[a], s);
    s = wave_sum(negA[a]); if (ln == 0) atomicAdd(&ws[3 + a], s);
    s = wave_sum(cntA[a]); if (ln == 0) atomicAdd(&ws[6 + a], s);
  }
}

// ---- Kernel 0: zero accumulators ------------------------------------------
__global__ void zero_ws_kernel(float* ws) {
  if (threadIdx.x < 32) ws[threadIdx.x] = 0.0f;
}

// ---- Kernel 2: gathers, smooth-L1 sums, mask counts -----------------------
__global__ void gather_l1_kernel(
    const float* __restrict__ embd_tl, const float* __restrict__ embd_br,
    const float* __restrict__ regs_tl, const float* __restrict__ regs_br,
    const float* __restrict__ regs_ct, const float* __restrict__ gt_tl,
    const float* __restrict__ gt_br, const float* __restrict__ gt_ct,
    const int* __restrict__ i_tl, const int* __restrict__ i_br,
    const int* __restrict__ i_ct, const int* __restrict__ masks,
    float* __restrict__ ws) {
  const int idx = blockIdx.x * blockDim.x + threadIdx.x;
  if (idx >= B_ * K_) return;
  const int b = idx >> 7;

  const float m = (float)masks[idx];
  const int itl = i_tl[idx], ibr = i_br[idx], ict = i_ct[idx];

  // embeddings (C == 1)
  const float e0 = embd_tl[b * HW_ + itl];
  const float e1 = embd_br[b * HW_ + ibr];
  ws[32 + idx]            = e0;
  ws[32 + B_ * K_ + idx]  = e1;

  float s_tl = 0.f, s_br = 0.f, s_ct = 0.f;
#pragma unroll
  for (int c = 0; c < 2; ++c) {
    float d, ad;
    d  = regs_tl[(b * 2 + c) * HW_ + itl] - gt_tl[idx * 2 + c];
    ad = fabsf(d);
    s_tl += (ad < 1.0f) ? 0.5f * d * d : ad - 0.5f;
    d  = regs_br[(b * 2 + c) * HW_ + ibr] - gt_br[idx * 2 + c];
    ad = fabsf(d);
    s_br += (ad < 1.0f) ? 0.5f * d * d : ad - 0.5f;
    d  = regs_ct[(b * 2 + c) * HW_ + ict] - gt_ct[idx * 2 + c];
    ad = fabsf(d);
    s_ct += (ad < 1.0f) ? 0.5f * d * d : ad - 0.5f;
  }
  atomicAdd(&ws[9],  m * s_tl);
  atomicAdd(&ws[10], m * s_br);
  atomicAdd(&ws[11], m * s_ct);
  atomicAdd(&ws[12], m);
  atomicAdd(&ws[16 + b], m);
}

// ---- Kernel 3: associative-embedding pull/push ----------------------------
__global__ void ae_kernel(const int* __restrict__ masks,
                          float* __restrict__ ws) {
  const int idx = blockIdx.x * blockDim.x + threadIdx.x;
  if (idx >= B_ * K_) return;
  const int b = idx >> 7;

  const float num  = ws[16 + b];
  const float mi   = (float)masks[idx];
  const float e0   = ws[32 + idx];
  const float e1   = ws[32 + B_ * K_ + idx];
  const float mean = 0.5f * (e0 + e1);

  const float d0 = e0 - mean, d1 = e1 - mean;
  const float pull = mi * (d0 * d0 + d1 * d1) / (num + EPSF);

  const float inv1 = 1.0f / (num + EPSF);
  const float inv2 = 1.0f / ((num - 1.0f) * num + EPSF);

  float acc = 0.f;
  const float* E0 = ws + 32 + b * K_;
  const float* E1 = ws + 32 + B_ * K_ + b * K_;
  const int* mrow = masks + b * K_;
  for (int j = 0; j < K_; ++j) {
    const float mj = (float)mrow[j];
    const float mean_j = 0.5f * (E0[j] + E1[j]);
    float dist = fmaxf(0.0f, 1.0f - fabsf(mean - mean_j));
    dist = (dist - inv1) * inv2;
    acc += mi * mj * dist;
  }
  atomicAdd(&ws[13], pull);
  atomicAdd(&ws[14], acc);
}

// ---- Kernel 4: final combine ----------------------------------------------
__global__ void combine_kernel(const float* __restrict__ ws,
                               float* __restrict__ out) {
  if (threadIdx.x == 0 && blockIdx.x == 0) {
    float focal = 0.f;
#pragma unroll
    for (int m = 0; m < 3; ++m) {
      const float pos = ws[m], neg = ws[3 + m], np = ws[6 + m];
      focal += (np > 0.f) ? -(pos + neg) / np : -neg;
    }
    const float num = ws[12] + EPSF;
    const float reg = (ws[9] + ws[10] + ws[11]) / num;
    out[0] = focal + 0.1f * (ws[13] + ws[14]) + reg;
  }
}

// ---------------------------------------------------------------------------
extern "C" void kernel_launch(void* const* d_in, const int* in_sizes, int n_in,
                              void* d_out, int out_size, void* d_ws,
                              size_t ws_size, hipStream_t stream) {
  const float* hmap_tl = (const float*)d_in[0];
  const float* hmap_br = (const float*)d_in[1];
  const float* hmap_ct = (const float*)d_in[2];
  const float* embd_tl = (const float*)d_in[3];
  const float* embd_br = (const float*)d_in[4];
  const float* regs_tl = (const float*)d_in[5];
  const float* regs_br = (const float*)d_in[6];
  const float* regs_ct = (const float*)d_in[7];
  const float* gt_tl   = (const float*)d_in[8];
  const float* gt_br   = (const float*)d_in[9];
  const float* gt_ct   = (const float*)d_in[10];
  const float* gtr_tl  = (const float*)d_in[11];
  const float* gtr_br  = (const float*)d_in[12];
  const float* gtr_ct  = (const float*)d_in[13];
  const int*   i_tl    = (const int*)d_in[14];
  const int*   i_br    = (const int*)d_in[15];
  const int*   i_ct    = (const int*)d_in[16];
  const int*   masks   = (const int*)d_in[17];

  float* ws  = (float*)d_ws;
  float* out = (float*)d_out;

  zero_ws_kernel<<<1, 32, 0, stream>>>(ws);
  focal_kernel<<<FOCAL_GRID, FOCAL_BLOCK, 0, stream>>>(
      hmap_tl, hmap_br, hmap_ct, gt_tl, gt_br, gt_ct, ws);
  gather_l1_kernel<<<(B_ * K_ + 255) / 256, 256, 0, stream>>>(
      embd_tl, embd_br, regs_tl, regs_br, regs_ct, gtr_tl, gtr_br, gtr_ct,
      i_tl, i_br, i_ct, masks, ws);
  ae_kernel<<<(B_ * K_ + 255) / 256, 256, 0, stream>>>(masks, ws);
  combine_kernel<<<1, 1, 0, stream>>>(ws, out);
}